// MultiModalGraphAttention_57312043597867
// MI455X (gfx1250) — compile-verified
//
#include <hip/hip_runtime.h>
#include <hip/hip_bf16.h>
#include <math.h>

#define N_NODES 16384
#define B_GRAPH 64
#define S_NODES 256
#define E_EDGES 524288
#define D_RAW   128
#define EA_DIM  32
#define ED_DIM  64
#define H_HEADS 16
#define C_CH    16
#define HIDDEN  256

typedef __attribute__((ext_vector_type(2))) float v2f;
typedef __attribute__((ext_vector_type(8))) float v8f;

// ---------------------------------------------------------------------------
// float atomic max via int/uint ordering trick (portable, no unsafe-fp-atomics)
// ---------------------------------------------------------------------------
__device__ __forceinline__ void atomic_max_float(float* addr, float val) {
  if (val >= 0.0f) {
    atomicMax((int*)addr, __float_as_int(val));
  } else {
    atomicMin((unsigned int*)addr, __float_as_uint(val));
  }
}

// ---------------------------------------------------------------------------
// 1. xn = x + sinusoidal positional encoding
// ---------------------------------------------------------------------------
__global__ void k_prep_xn(const float* __restrict__ x, float* __restrict__ xn) {
  int idx = blockIdx.x * blockDim.x + threadIdx.x;
  if (idx >= N_NODES * D_RAW) return;
  int pos = idx / D_RAW, d = idx % D_RAW;
  const float kf = -logf(10000.0f) / (float)D_RAW;
  float div = expf((float)((d >> 1) * 2) * kf);
  float ang = (float)pos * div;
  float pe = (d & 1) ? cosf(ang) : sinf(ang);
  xn[idx] = x[idx] + pe;
}

// ---------------------------------------------------------------------------
// Generic f32 WMMA GEMM: C[M,N] = (relu?)(A[M,K] @ W[K,N] + bias)
// One wave per 16x16 output tile, V_WMMA_F32_16X16X4_F32 over K.
// A-frag: lane = m + 16*(k>=2), vgprs = k%2.  B-frag: lane = n + 16*(k>=2).
// C-frag: vgpr r -> row r (lanes<16) / r+8 (lanes>=16), col = lane%16.
// ---------------------------------------------------------------------------
template <int RELU>
__global__ void k_gemm(const float* __restrict__ A, const float* __restrict__ W,
                       const float* __restrict__ bias, float* __restrict__ C,
                       int M, int K, int N) {
  const int lane = threadIdx.x;
  const int row0 = (blockIdx.x * blockDim.y + threadIdx.y) * 16;
  const int col0 = blockIdx.y * 16;
  if (row0 >= M) return;
  const int l16 = lane & 15;
  const int kh  = (lane >> 4) << 1;   // 0 or 2
  const int rb  = (lane >> 4) * 8;    // row base for C frag
  v8f acc = {};
  const float* arow = A + (size_t)(row0 + l16) * K;
  for (int k0 = 0; k0 < K; k0 += 4) {
    int ka = k0 + kh;
    v2f a, b;
    a.x = arow[ka];
    a.y = arow[ka + 1];
    b.x = W[(size_t)ka * N + col0 + l16];
    b.y = W[(size_t)(ka + 1) * N + col0 + l16];
    acc = __builtin_amdgcn_wmma_f32_16x16x4_f32(false, a, false, b, (short)0, acc,
                                                false, false);
  }
  float bv = bias ? bias[col0 + l16] : 0.0f;
  for (int r = 0; r < 8; ++r) {
    float v = acc[r] + bv;
    if (RELU) v = fmaxf(v, 0.0f);
    C[(size_t)(row0 + rb + r) * N + col0 + l16] = v;
  }
}

// ---------------------------------------------------------------------------
// 3. per-node attention logits: a_src/a_dst[n,h] = sum_c h[n,h,c]*att[h,c]
// ---------------------------------------------------------------------------
__global__ void k_node_att(const float* __restrict__ h, const float* __restrict__ att_src,
                           const float* __restrict__ att_dst, float* __restrict__ a_src,
                           float* __restrict__ a_dst) {
  int idx = blockIdx.x * blockDim.x + threadIdx.x;
  if (idx >= N_NODES * H_HEADS) return;
  int n = idx >> 4, hd = idx & 15;
  const float* hp = h + (size_t)n * HIDDEN + hd * C_CH;
  float s = 0.f, d = 0.f;
  for (int c = 0; c < C_CH; ++c) {
    float hv = hp[c];
    s += hv * att_src[hd * C_CH + c];
    d += hv * att_dst[hd * C_CH + c];
  }
  a_src[idx] = s;
  a_dst[idx] = d;
}

// ---------------------------------------------------------------------------
// 4. Ae[d,h] = sum_c W_gat_edge[d, h*16+c] * att_edge[h,c]   (64x16)
// ---------------------------------------------------------------------------
__global__ void k_make_ae(const float* __restrict__ W_gat_edge,
                          const float* __restrict__ att_edge, float* __restrict__ Ae) {
  int idx = blockIdx.x * blockDim.x + threadIdx.x;
  if (idx >= ED_DIM * H_HEADS) return;
  int d = idx >> 4, hd = idx & 15;
  float s = 0.f;
  for (int c = 0; c < C_CH; ++c)
    s += W_gat_edge[(size_t)d * HIDDEN + hd * C_CH + c] * att_edge[hd * C_CH + c];
  Ae[idx] = s;
}

// ---------------------------------------------------------------------------
// 5. init accumulators (ws is poisoned by harness; must be re-inited each call)
// ---------------------------------------------------------------------------
__global__ void k_init(float* __restrict__ amax, float* __restrict__ denom,
                       float* __restrict__ gat) {
  int idx = blockIdx.x * blockDim.x + threadIdx.x;
  if (idx < N_NODES * H_HEADS) {
    amax[idx] = -__builtin_inff();
    denom[idx] = 0.0f;
  }
  if (idx < N_NODES * HIDDEN) gat[idx] = 0.0f;
}

// ---------------------------------------------------------------------------
// 6. fused edge pipeline: ef = relu(edge_attr@W_edge+b) (WMMA) -> LDS ->
//    a_e = ef@Ae (WMMA) -> alpha = leaky_relu(a_src[src]+a_dst[dst]+a_e)
//    -> store alpha, atomic-max into amax[dst]
//    one wave handles 16 edges; 4 waves per block.
// ---------------------------------------------------------------------------
__global__ void k_edge_alpha(const float* __restrict__ edge_attr,
                             const float* __restrict__ W_edge,
                             const float* __restrict__ b_edge,
                             const float* __restrict__ Ae,
                             const int* __restrict__ srcI, const int* __restrict__ dstI,
                             const float* __restrict__ a_src,
                             const float* __restrict__ a_dst,
                             float* __restrict__ alpha, float* __restrict__ amax) {
  __shared__ float lds[4][16][ED_DIM + 4];
  const int lane = threadIdx.x, wv = threadIdx.y;
  const int e0 = (blockIdx.x * 4 + wv) * 16;
  const int l16 = lane & 15;
  const int kh  = (lane >> 4) << 1;
  const int rb  = (lane >> 4) * 8;
  // stage 1: ef tile [16 edges x 64]
  const float* arow = edge_attr + (size_t)(e0 + l16) * EA_DIM;
  for (int ct = 0; ct < 4; ++ct) {
    v8f acc = {};
    for (int k0 = 0; k0 < EA_DIM; k0 += 4) {
      int ka = k0 + kh;
      v2f a, b;
      a.x = arow[ka];
      a.y = arow[ka + 1];
      b.x = W_edge[(size_t)ka * ED_DIM + ct * 16 + l16];
      b.y = W_edge[(size_t)(ka + 1) * ED_DIM + ct * 16 + l16];
      acc = __builtin_amdgcn_wmma_f32_16x16x4_f32(false, a, false, b, (short)0, acc,
                                                  false, false);
    }
    float bv = b_edge[ct * 16 + l16];
    for (int r = 0; r < 8; ++r)
      lds[wv][rb + r][ct * 16 + l16] = fmaxf(acc[r] + bv, 0.0f);
  }
  __syncthreads();
  // stage 2: a_e = ef @ Ae  [16 x 16]
  v8f acc2 = {};
  for (int k0 = 0; k0 < ED_DIM; k0 += 4) {
    int ka = k0 + kh;
    v2f a, b;
    a.x = lds[wv][l16][ka];
    a.y = lds[wv][l16][ka + 1];
    b.x = Ae[ka * H_HEADS + l16];
    b.y = Ae[(ka + 1) * H_HEADS + l16];
    acc2 = __builtin_amdgcn_wmma_f32_16x16x4_f32(false, a, false, b, (short)0, acc2,
                                                 false, false);
  }
  // epilogue: gather + leaky_relu + atomic max
  const int hd = l16;
  for (int r = 0; r < 8; ++r) {
    int e = e0 + rb + r;
    int s = srcI[e], d = dstI[e];
    float al = a_src[s * H_HEADS + hd] + a_dst[d * H_HEADS + hd] + acc2[r];
    al = (al > 0.0f) ? al : 0.2f * al;
    alpha[(size_t)e * H_HEADS + hd] = al;
    atomic_max_float(&amax[d * H_HEADS + hd], al);
  }
}

// 7. amax = where(isfinite, amax, 0)
__global__ void k_fix_amax(float* __restrict__ amax) {
  int idx = blockIdx.x * blockDim.x + threadIdx.x;
  if (idx >= N_NODES * H_HEADS) return;
  float v = amax[idx];
  if (v < -3.0e38f) amax[idx] = 0.0f;
}

// 8. ex = exp(alpha - amax[dst]); denom[dst] += ex
__global__ void k_exp(const int* __restrict__ dstI, const float* __restrict__ amax,
                      float* __restrict__ alpha, float* __restrict__ denom) {
  int idx = blockIdx.x * blockDim.x + threadIdx.x;
  if (idx >= E_EDGES * H_HEADS) return;
  int e = idx >> 4, hd = idx & 15;
  int d = dstI[e];
  float ex = expf(alpha[idx] - amax[d * H_HEADS + hd]);
  alpha[idx] = ex;
  atomicAdd(&denom[d * H_HEADS + hd], ex);
}

// 9. gat[dst] += (ex/denom[dst]) * h[src]
__global__ void k_scatter(const int* __restrict__ srcI, const int* __restrict__ dstI,
                          const float* __restrict__ alpha, const float* __restrict__ denom,
                          const float* __restrict__ hbuf, float* __restrict__ gat) {
  int idx = blockIdx.x * blockDim.x + threadIdx.x;
  if (idx >= E_EDGES * H_HEADS) return;
  int e = idx >> 4, hd = idx & 15;
  int s = srcI[e], d = dstI[e];
  float w = alpha[idx] / (denom[d * H_HEADS + hd] + 1e-16f);
  const float* hp = hbuf + (size_t)s * HIDDEN + hd * C_CH;
  float* gp = gat + (size_t)d * HIDDEN + hd * C_CH;
  for (int c = 0; c < C_CH; ++c) atomicAdd(&gp[c], w * hp[c]);
}

// 10. local = relu(gat + b_gat) in place
__global__ void k_bias_relu(float* __restrict__ gat, const float* __restrict__ b_gat) {
  int idx = blockIdx.x * blockDim.x + threadIdx.x;
  if (idx >= N_NODES * HIDDEN) return;
  gat[idx] = fmaxf(gat[idx] + b_gat[idx & (HIDDEN - 1)], 0.0f);
}

// ---------------------------------------------------------------------------
// 12. MHA per (graph, head, 16-row strip): scores (WMMA) -> LDS softmax ->
//     p @ v (WMMA).  Never materializes full score tensor.
// ---------------------------------------------------------------------------
__global__ void k_attn(const float* __restrict__ q, const float* __restrict__ k,
                       const float* __restrict__ v, float* __restrict__ ao) {
  __shared__ float sp[16][S_NODES + 4];
  const int b  = blockIdx.x >> 4;   // S/16 = 16 row tiles per graph
  const int rt = blockIdx.x & 15;
  const int hd = blockIdx.y;
  const int lane = threadIdx.x;
  const int l16 = lane & 15;
  const int kh  = (lane >> 4) << 1;
  const int rb  = (lane >> 4) * 8;
  const int rbase = b * S_NODES + rt * 16;
  const int cb0 = b * S_NODES;
  const int hoff = hd * C_CH;
  // stage A: score strip [16 x 256], scale 1/sqrt(16)
  const float* qrow = q + (size_t)(rbase + l16) * HIDDEN + hoff;
  for (int ct = 0; ct < 16; ++ct) {
    v8f acc = {};
    const float* kcol = k + (size_t)(cb0 + ct * 16 + l16) * HIDDEN + hoff;
    for (int ks = 0; ks < C_CH; ks += 4) {
      int ka = ks + kh;
      v2f a, bb;
      a.x = qrow[ka];
      a.y = qrow[ka + 1];
      bb.x = kcol[ka];
      bb.y = kcol[ka + 1];
      acc = __builtin_amdgcn_wmma_f32_16x16x4_f32(false, a, false, bb, (short)0, acc,
                                                  false, false);
    }
    for (int r = 0; r < 8; ++r) sp[rb + r][ct * 16 + l16] = acc[r] * 0.25f;
  }
  __syncthreads();
  // stage B: softmax over each of the 16 rows (256 cols)
  if (lane < 16) {
    float mx = -__builtin_inff();
    for (int j = 0; j < S_NODES; ++j) mx = fmaxf(mx, sp[lane][j]);
    float sm = 0.0f;
    for (int j = 0; j < S_NODES; ++j) {
      float e_ = expf(sp[lane][j] - mx);
      sp[lane][j] = e_;
      sm += e_;
    }
    float inv = 1.0f / sm;
    for (int j = 0; j < S_NODES; ++j) sp[lane][j] *= inv;
  }
  __syncthreads();
  // stage C: out[16 x 16] = p[16 x 256] @ v[256 x 16]
  v8f acc2 = {};
  for (int k0 = 0; k0 < S_NODES; k0 += 4) {
    int ka = k0 + kh;
    v2f a, bb;
    a.x = sp[l16][ka];
    a.y = sp[l16][ka + 1];
    bb.x = v[(size_t)(cb0 + ka) * HIDDEN + hoff + l16];
    bb.y = v[(size_t)(cb0 + ka + 1) * HIDDEN + hoff + l16];
    acc2 = __builtin_amdgcn_wmma_f32_16x16x4_f32(false, a, false, bb, (short)0, acc2,
                                                 false, false);
  }
  for (int r = 0; r < 8; ++r)
    ao[(size_t)(rbase + rb + r) * HIDDEN + hoff + l16] = acc2[r];
}

// 14. mean pool over S nodes per graph
__global__ void k_pool(const float* __restrict__ gout, float* __restrict__ pooled) {
  int idx = blockIdx.x * blockDim.x + threadIdx.x;
  if (idx >= B_GRAPH * HIDDEN) return;
  int b = idx >> 8, ch = idx & (HIDDEN - 1);
  const float* p = gout + (size_t)b * S_NODES * HIDDEN + ch;
  float s = 0.0f;
  for (int j = 0; j < S_NODES; ++j) s += p[(size_t)j * HIDDEN];
  pooled[idx] = s * (1.0f / S_NODES);
}

// 16. out[b] = hdn[b,:] @ W2 + b2
__global__ void k_final(const float* __restrict__ hdn, const float* __restrict__ W2,
                        const float* __restrict__ b2, float* __restrict__ out) {
  int b = threadIdx.x;
  if (b >= B_GRAPH) return;
  float s = 0.0f;
  for (int c = 0; c < HIDDEN; ++c) s += hdn[b * HIDDEN + c] * W2[c];
  out[b] = s + b2[0];
}

// ---------------------------------------------------------------------------
extern "C" void kernel_launch(void* const* d_in, const int* in_sizes, int n_in,
                              void* d_out, int out_size, void* d_ws, size_t ws_size,
                              hipStream_t stream) {
  (void)in_sizes; (void)n_in; (void)out_size; (void)ws_size;
  const float* x          = (const float*)d_in[0];
  const int*   ei         = (const int*)d_in[1];
  const float* edge_attr  = (const float*)d_in[2];
  const float* W_edge     = (const float*)d_in[3];
  const float* b_edge     = (const float*)d_in[4];
  const float* W_gat      = (const float*)d_in[5];
  const float* b_gat      = (const float*)d_in[6];
  const float* W_gat_edge = (const float*)d_in[7];
  const float* att_src    = (const float*)d_in[8];
  const float* att_dst    = (const float*)d_in[9];
  const float* att_edge   = (const float*)d_in[10];
  const float* Wq = (const float*)d_in[11]; const float* bq = (const float*)d_in[12];
  const float* Wk = (const float*)d_in[13]; const float* bk = (const float*)d_in[14];
  const float* Wv = (const float*)d_in[15]; const float* bv = (const float*)d_in[16];
  const float* Wo = (const float*)d_in[17]; const float* bo = (const float*)d_in[18];
  const float* W1 = (const float*)d_in[19]; const float* b1 = (const float*)d_in[20];
  const float* W2 = (const float*)d_in[21]; const float* b2 = (const float*)d_in[22];
  const int* srcI = ei;
  const int* dstI = ei + E_EDGES;

  char* wsp = (char*)d_ws;
  size_t off = 0;
  auto carve = [&](size_t bytes) -> float* {
    float* p = (float*)(wsp + off);
    off += (bytes + 255) & ~(size_t)255;
    return p;
  };
  float* xn     = carve((size_t)N_NODES * D_RAW * 4);
  float* hbuf   = carve((size_t)N_NODES * HIDDEN * 4);
  float* a_src  = carve((size_t)N_NODES * H_HEADS * 4);
  float* a_dst  = carve((size_t)N_NODES * H_HEADS * 4);
  float* Ae     = carve((size_t)ED_DIM * H_HEADS * 4);
  float* alpha  = carve((size_t)E_EDGES * H_HEADS * 4);
  float* amax   = carve((size_t)N_NODES * H_HEADS * 4);
  float* denom  = carve((size_t)N_NODES * H_HEADS * 4);
  float* gat    = carve((size_t)N_NODES * HIDDEN * 4);
  float* qb     = carve((size_t)N_NODES * HIDDEN * 4);
  float* kb     = carve((size_t)N_NODES * HIDDEN * 4);
  float* vb     = carve((size_t)N_NODES * HIDDEN * 4);
  float* ao     = carve((size_t)N_NODES * HIDDEN * 4);
  float* pooled = carve((size_t)B_GRAPH * HIDDEN * 4);
  float* hdn    = carve((size_t)B_GRAPH * HIDDEN * 4);

  // 1. positional encoding
  k_prep_xn<<<(N_NODES * D_RAW + 255) / 256, 256, 0, stream>>>(x, xn);
  // 2. h = xn @ W_gat  (no bias here; b_gat added post-aggregation)
  dim3 gb(32, 4);
  k_gemm<0><<<dim3(N_NODES / 64, HIDDEN / 16), gb, 0, stream>>>(xn, W_gat, nullptr, hbuf,
                                                                N_NODES, D_RAW, HIDDEN);
  // 3. per-node attention logits
  k_node_att<<<(N_NODES * H_HEADS + 255) / 256, 256, 0, stream>>>(hbuf, att_src, att_dst,
                                                                  a_src, a_dst);
  // 4. collapsed edge-attention matrix Ae
  k_make_ae<<<(ED_DIM * H_HEADS + 255) / 256, 256, 0, stream>>>(W_gat_edge, att_edge, Ae);
  // 5. init accumulators
  k_init<<<(N_NODES * HIDDEN + 255) / 256, 256, 0, stream>>>(amax, denom, gat);
  // 6. fused edge MLP + a_e GEMM + alpha + segment max
  k_edge_alpha<<<E_EDGES / 64, gb, 0, stream>>>(edge_attr, W_edge, b_edge, Ae, srcI, dstI,
                                                a_src, a_dst, alpha, amax);
  // 7-9. edge softmax + scatter
  k_fix_amax<<<(N_NODES * H_HEADS + 255) / 256, 256, 0, stream>>>(amax);
  k_exp<<<(E_EDGES * H_HEADS + 255) / 256, 256, 0, stream>>>(dstI, amax, alpha, denom);
  k_scatter<<<(E_EDGES * H_HEADS + 255) / 256, 256, 0, stream>>>(srcI, dstI, alpha, denom,
                                                                 hbuf, gat);
  // 10. local = relu(gat + b_gat)
  k_bias_relu<<<(N_NODES * HIDDEN + 255) / 256, 256, 0, stream>>>(gat, b_gat);
  // 11. q, k, v projections
  dim3 gg(N_NODES / 64, HIDDEN / 16);
  k_gemm<0><<<gg, gb, 0, stream>>>(gat, Wq, bq, qb, N_NODES, HIDDEN, HIDDEN);
  k_gemm<0><<<gg, gb, 0, stream>>>(gat, Wk, bk, kb, N_NODES, HIDDEN, HIDDEN);
  k_gemm<0><<<gg, gb, 0, stream>>>(gat, Wv, bv, vb, N_NODES, HIDDEN, HIDDEN);
  // 12. multi-head self-attention
  k_attn<<<dim3(B_GRAPH * (S_NODES / 16), H_HEADS), dim3(32), 0, stream>>>(qb, kb, vb, ao);
  // 13. output projection (reuse qb as gout)
  k_gemm<0><<<gg, gb, 0, stream>>>(ao, Wo, bo, qb, N_NODES, HIDDEN, HIDDEN);
  // 14. mean pool
  k_pool<<<(B_GRAPH * HIDDEN + 255) / 256, 256, 0, stream>>>(qb, pooled);
  // 15. hdn = relu(pooled @ W1 + b1)
  k_gemm<1><<<dim3(1, HIDDEN / 16), gb, 0, stream>>>(pooled, W1, b1, hdn, B_GRAPH, HIDDEN,
                                                     HIDDEN);
  // 16. final projection -> [B,1]
  k_final<<<1, 64, 0, stream>>>(hdn, W2, b2, (float*)d_out);
}